// PaiNNLayer_84576495993158
// MI455X (gfx1250) — compile-verified
//
#include <hip/hip_runtime.h>
#include <math.h>

// ---------------------------------------------------------------------------
// PaiNN layer on gfx1250 (wave32, WMMA), fp32 end-to-end with
// V_WMMA_F32_16X16X4_F32. One wave owns a 16x16 f32 tile, K stepped by 4.
//
// v2 changes vs v1:
//  * all ragged weights zero-padded + pair-interleaved in ws once per call
//    -> branch-free GEMM inner loop, b64 fragment loads
//  * edge kernel keeps packed Ww in LDS (48KB) behind a cooperative fill
//  * h-MLP still evaluated per node (2048+512 rows), gathered per edge
// ---------------------------------------------------------------------------

typedef float  v2f __attribute__((ext_vector_type(2)));
typedef float  v8f __attribute__((ext_vector_type(8)));

// Fragment layouts (CDNA5 ISA 7.12.2, 32-bit, wave32):
//   A 16x4 : lane m=lane&15; half0 holds K=k,k+1 ; half1 holds K=k+2,k+3
//   B 4x16 : lane n=lane&15; half0 holds rows k,k+1 ; half1 rows k+2,k+3
//   C 16x16: VGPR r -> row (half*8 + r), col = lane&15
__device__ __forceinline__ v8f wmma_f32_k4(v2f a, v2f b, v8f c) {
    return __builtin_amdgcn_wmma_f32_16x16x4_f32(false, a, false, b, (short)0, c,
                                                 false, false);
}

__device__ __forceinline__ float silu(float x) {
    return x * (1.0f / (1.0f + __expf(-x)));
}

// ---------------------------------------------------------------------------
// Weight repacking: B[k][n] (real dims Kr x Nr) -> Bp[(k>>1)*(2*Npad) + 2n + (k&1)]
// zero-padded to Kpad x Npad.  Makes each B WMMA fragment one aligned b64 load.
// ---------------------------------------------------------------------------
__global__ void pack_pairs(const float* __restrict__ B, int Kr, int Nr,
                           float* __restrict__ Bp, int Kpad, int Npad)
{
    const int idx = blockIdx.x * blockDim.x + threadIdx.x;
    if (idx >= Kpad * Npad) return;
    const int k = idx / Npad;
    const int n = idx - k * Npad;
    const float v = (k < Kr && n < Nr) ? B[k * Nr + n] : 0.0f;
    Bp[(k >> 1) * (2 * Npad) + 2 * n + (k & 1)] = v;
}

__global__ void pad_vec(const float* __restrict__ b, int Nr,
                        float* __restrict__ bp, int Npad)
{
    const int i = blockIdx.x * blockDim.x + threadIdx.x;
    if (i < Npad) bp[i] = (i < Nr) ? b[i] : 0.0f;
}

// ---------------------------------------------------------------------------
// fp32 WMMA GEMM, branch-free inner loop.
//   C[m,n] = act( sum_k A[m,k]*B[k,n] + bias[n] )
//   A elem addr = A + m*lda + k*S        (S=3 for [N,D,3] spatial tensors)
//   C elem addr = C + m*ldc + n*S
//   Bp is pair-interleaved [Kpad/2][Npad][2]; bias pre-padded (or null).
//   blockIdx.z offsets A and C by the spatial component (0..2).
// ---------------------------------------------------------------------------
template <int S>
__global__ __launch_bounds__(256) void gemm16(
    const float* __restrict__ A, int lda,
    const float* __restrict__ Bp, int Npad,
    const float* __restrict__ bias,
    float* __restrict__ C, int ldc,
    int M, int K, int act)
{
    A += blockIdx.z;
    C += blockIdx.z;
    const int wave  = threadIdx.x >> 5;
    const int lane  = threadIdx.x & 31;
    const int nTn   = Npad >> 4;
    const int tiles = (M >> 4) * nTn;
    const int tile  = blockIdx.x * 8 + wave;
    if (tile >= tiles) return;
    const int tm   = tile / nTn;
    const int tn   = tile - tm * nTn;
    const int half = lane >> 4;
    const int l16  = lane & 15;
    const int col  = tn * 16 + l16;

    const float bv = bias ? bias[col] : 0.0f;
    v8f acc;
#pragma unroll
    for (int r = 0; r < 8; ++r) acc[r] = bv;

    const int arow  = (tm * 16 + l16) * lda;
    const int N2    = 2 * Npad;
    const int col2  = 2 * col;
    const int half2 = half * 2;

#pragma unroll 4
    for (int k = 0; k < K; k += 4) {
        const int ka = k + half2;
        v2f a;
        if (S == 1) {
            a = *(const v2f*)(A + arow + ka);
        } else {
            a.x = A[arow + ka * S];
            a.y = A[arow + (ka + 1) * S];
        }
        const v2f b = *(const v2f*)(Bp + ((k >> 1) + half) * N2 + col2);
        acc = wmma_f32_k4(a, b, acc);
    }

#pragma unroll
    for (int r = 0; r < 8; ++r) {
        float v = acc[r];
        if (act) v = silu(v);
        const int m = tm * 16 + half * 8 + r;
        C[m * ldc + col * S] = v;
    }
}

// ---------------------------------------------------------------------------
// Fused edge kernel for one edge type. One wave handles 16 consecutive edges:
//   we  = dist[16,32] @ Ww[32,384] + bw        (WMMA, K=32, Ww in LDS)
//   phi = we * hx[send]                        (gather, fused in C layout)
//   z_s[recv] += phi_s
//   z_v[recv] += phi_vv * send_v[send] + phi_vs * dir   (3 atomics per (e,d))
// ---------------------------------------------------------------------------
__global__ __launch_bounds__(256) void edge_kernel(
    const float* __restrict__ dist,   // [E,32]
    const float* __restrict__ dirs,   // [E,3]
    const int*   __restrict__ send,   // [E]
    const int*   __restrict__ recv,   // [E]
    const float* __restrict__ Wwp,    // packed [16][384][2] = 12288 floats
    const float* __restrict__ bw,     // [384]
    const float* __restrict__ hx,     // [Nsend,384]  (precomputed h(send_s))
    const float* __restrict__ sv,     // [Nsend,128,3]
    float* __restrict__ z_s,          // [Nrecv,128]
    float* __restrict__ z_v,          // [Nrecv,128,3]
    int E)
{
    __shared__ float lw[12288];       // 48KB: packed Ww, reused by all 8 waves
    {
        float4*       dst = (float4*)lw;
        const float4* src = (const float4*)Wwp;
#pragma unroll
        for (int i = 0; i < 12; ++i)               // 3072 float4 / 256 threads
            dst[threadIdx.x + 256 * i] = src[threadIdx.x + 256 * i];
    }
    __syncthreads();

    const int wave = threadIdx.x >> 5;
    const int lane = threadIdx.x & 31;
    const int e0 = (blockIdx.x * 8 + wave) * 16;
    if (e0 >= E) return;
    const int half = lane >> 4;
    const int l16  = lane & 15;

    // A fragments for all 8 K-steps: dist rows are contiguous edges -> b64 loads
    v2f a[8];
#pragma unroll
    for (int ks = 0; ks < 8; ++ks)
        a[ks] = *(const v2f*)(dist + (e0 + l16) * 32 + ks * 4 + half * 2);

    int si[8], ri[8];
#pragma unroll
    for (int r = 0; r < 8; ++r) {
        const int e = e0 + half * 8 + r;   // row owned by this lane in C layout
        si[r] = send[e];
        ri[r] = recv[e];
    }

    // ---- scalar part: cols 0..127 of we ----
    for (int t = 0; t < 8; ++t) {
        const int col = t * 16 + l16;
        v8f acc;
        const float bv = bw[col];
#pragma unroll
        for (int r = 0; r < 8; ++r) acc[r] = bv;
#pragma unroll
        for (int ks = 0; ks < 8; ++ks) {
            const v2f b = *(const v2f*)(lw + (ks * 2 + half) * 768 + 2 * col);
            acc = wmma_f32_k4(a[ks], b, acc);
        }
#pragma unroll
        for (int r = 0; r < 8; ++r) {
            const float phi = acc[r] * hx[si[r] * 384 + col];
            atomicAdd(&z_s[ri[r] * 128 + col], phi);
        }
    }

    // ---- vector part: cols 128..255 (vv gate) and 256..383 (vs gate) ----
    for (int t = 0; t < 8; ++t) {
        const int d    = t * 16 + l16;
        const int colv = 128 + d;
        const int cols = 256 + d;
        v8f avv, avs;
        const float bvv = bw[colv];
        const float bvs = bw[cols];
#pragma unroll
        for (int r = 0; r < 8; ++r) { avv[r] = bvv; avs[r] = bvs; }
#pragma unroll
        for (int ks = 0; ks < 8; ++ks) {
            const v2f b1 = *(const v2f*)(lw + (ks * 2 + half) * 768 + 2 * colv);
            const v2f b2 = *(const v2f*)(lw + (ks * 2 + half) * 768 + 2 * cols);
            avv = wmma_f32_k4(a[ks], b1, avv);
            avs = wmma_f32_k4(a[ks], b2, avs);
        }
#pragma unroll
        for (int r = 0; r < 8; ++r) {
            const float pvv = avv[r] * hx[si[r] * 384 + colv];
            const float pvs = avs[r] * hx[si[r] * 384 + cols];
            const int   e   = e0 + half * 8 + r;
            const float* dp  = &dirs[e * 3];
            const float* svp = &sv[(si[r] * 128 + d) * 3];
            float* zp = &z_v[(ri[r] * 128 + d) * 3];
            atomicAdd(zp + 0, pvv * svp[0] + pvs * dp[0]);
            atomicAdd(zp + 1, pvv * svp[1] + pvs * dp[1]);
            atomicAdd(zp + 2, pvv * svp[2] + pvs * dp[2]);
        }
    }
}

// gcat[n] = concat( z_s[n], ||Vv[n,:,*]||_2 )  -> [N,256]
__global__ void norm_concat(const float* __restrict__ z_s,
                            const float* __restrict__ Vv,
                            float* __restrict__ gcat, int Nrecv)
{
    const int idx = blockIdx.x * blockDim.x + threadIdx.x;
    if (idx >= Nrecv * 128) return;
    const int n = idx >> 7, d = idx & 127;
    const float* p = &Vv[(n * 128 + d) * 3];
    const float nm = sqrtf(p[0] * p[0] + p[1] * p[1] + p[2] * p[2]);
    gcat[n * 256 + d]       = z_s[idx];
    gcat[n * 256 + 128 + d] = nm;
}

// out_s += a_sv * <Uv,Vv> + a_ss ;  out_v += Uv * a_vv
__global__ void finalize_add(const float* __restrict__ g,
                             const float* __restrict__ Uv,
                             const float* __restrict__ Vv,
                             float* __restrict__ out_s,
                             float* __restrict__ out_v, int Nrecv)
{
    const int idx = blockIdx.x * blockDim.x + threadIdx.x;
    if (idx >= Nrecv * 128) return;
    const int n = idx >> 7, d = idx & 127;
    const float a_ss = g[n * 384 + d];
    const float a_vv = g[n * 384 + 128 + d];
    const float a_sv = g[n * 384 + 256 + d];
    const float* up = &Uv[(n * 128 + d) * 3];
    const float* vp = &Vv[(n * 128 + d) * 3];
    const float dot = up[0] * vp[0] + up[1] * vp[1] + up[2] * vp[2];
    out_s[idx] += a_sv * dot + a_ss;
    float* ov = &out_v[idx * 3];
    ov[0] += up[0] * a_vv;
    ov[1] += up[1] * a_vv;
    ov[2] += up[2] * a_vv;
}

// ---------------------------------------------------------------------------
// Host-side launch
// ---------------------------------------------------------------------------
extern "C" void kernel_launch(void* const* d_in, const int* in_sizes, int n_in,
                              void* d_out, int out_size, void* d_ws, size_t ws_size,
                              hipStream_t stream)
{
    (void)n_in; (void)out_size; (void)ws_size;
    constexpr int D = 128, NE = 2048, NN = 512;
    float* out = (float*)d_out;
    float* ws  = (float*)d_ws;

    // workspace layout (float offsets); total 6,394,432 floats (~25.6 MB)
    float* hx_e  = ws;               // [2048,384]
    float* hx_n  = ws + 786432;      // [512,384]
    float* t1    = ws + 983040;      // [2048,224]  h hidden (reused)
    float* zs    = ws + 1441792;     // [2048,128]
    float* zv    = ws + 1703936;     // [2048,128,3]
    float* Vv    = ws + 2490368;     // [2048,128,3]
    float* Uv    = ws + 3276800;     // [2048,128,3]
    float* gcat  = ws + 4063232;     // [2048,256]
    float* t2    = ws + 4587520;     // [2048,320]  g hidden
    float* gout  = ws + 5242880;     // [2048,384]
    float* Wh1p  = ws + 6029312;     // packed 128x224
    float* bh1p  = ws + 6057984;     // 224
    float* Wh2p  = ws + 6058240;     // packed 224x384
    float* Wwp   = ws + 6144256;     // packed 32x384   (per type, reused)
    float* Vp    = ws + 6156544;     // packed 128x128
    float* Up    = ws + 6172928;     // packed 128x128
    float* Wg1p  = ws + 6189312;     // packed 256x320
    float* bg1p  = ws + 6271232;     // 320
    float* Wg2p  = ws + 6271552;     // packed 320x384  -> end 6394432

    const float* elec_s = (const float*)d_in[0];
    const float* elec_v = (const float*)d_in[1];
    const float* nuc_s  = (const float*)d_in[2];
    const float* nuc_v  = (const float*)d_in[3];

    // out = inputs (updates are added per edge type afterwards)
    hipMemcpyAsync(out,           elec_s, (size_t)NE * D * 4,     hipMemcpyDeviceToDevice, stream);
    hipMemcpyAsync(out + 262144,  elec_v, (size_t)NE * D * 3 * 4, hipMemcpyDeviceToDevice, stream);
    hipMemcpyAsync(out + 1048576, nuc_s,  (size_t)NN * D * 4,     hipMemcpyDeviceToDevice, stream);
    hipMemcpyAsync(out + 1114112, nuc_v,  (size_t)NN * D * 3 * 4, hipMemcpyDeviceToDevice, stream);

    auto pack = [&](const float* B, int Kr, int Nr, float* Bp, int Kp, int Np) {
        pack_pairs<<<(Kp * Np + 255) / 256, 256, 0, stream>>>(B, Kr, Nr, Bp, Kp, Np);
    };
    auto gemmD = [&](const float* A, int lda, const float* Bp, int Np,
                     const float* bias, float* C, int ldc, int M, int K, int act) {
        const int tiles = (M / 16) * (Np / 16);
        gemm16<1><<<dim3((tiles + 7) / 8, 1, 1), 256, 0, stream>>>(
            A, lda, Bp, Np, bias, C, ldc, M, K, act);
    };
    auto gemmV = [&](const float* A, const float* Bp, float* C, int M) {
        const int tiles = (M / 16) * 8;
        gemm16<3><<<dim3((tiles + 7) / 8, 1, 3), 256, 0, stream>>>(
            A, 384, Bp, 128, nullptr, C, 384, M, 128, 0);
    };

    // --- params (jax pytree leaf order: keys sorted U,V,g,h,w; labels sorted) ---
    // U: 24..28 (anti,en,ne,nn,same)   V: 29..33
    // g: 34+4*idx -> {W1,b1,W2,b2}     h: 54..57   w: 58+2*idx -> {W,b}
    const float* Wh1 = (const float*)d_in[54];  // [128,222]
    const float* bh1 = (const float*)d_in[55];
    const float* Wh2 = (const float*)d_in[56];  // [222,384]
    const float* bh2 = (const float*)d_in[57];  // [384]

    pack(Wh1, 128, 222, Wh1p, 128, 224);
    pad_vec<<<1, 224, 0, stream>>>(bh1, 222, bh1p, 224);
    pack(Wh2, 222, 384, Wh2p, 224, 384);

    // h-MLP once per node set (the big algebraic saving vs per-edge h)
    gemmD(elec_s, 128, Wh1p, 224, bh1p, t1, 224, NE, 128, 1);
    gemmD(t1, 224, Wh2p, 384, bh2, hx_e, 384, NE, 224, 0);
    gemmD(nuc_s, 128, Wh1p, 224, bh1p, t1, 224, NN, 128, 1);
    gemmD(t1, 224, Wh2p, 384, bh2, hx_n, 384, NN, 224, 0);

    // per-type descriptors: input base, g/U/V/w leaf indices, topology
    struct TD { int in, g, U, V, w; const float* hx; const float* sv; int Nr; int outS; int outV; };
    const TD tds[5] = {
        // label  in   g   U   V   w    hx    send_v  Nrecv outS     outV
        /*same*/ { 4, 50, 28, 33, 66, hx_e, elec_v, NE, 0,       262144 },
        /*anti*/ { 8, 34, 24, 29, 58, hx_e, elec_v, NE, 0,       262144 },
        /*ne  */ {12, 42, 26, 31, 62, hx_n, nuc_v,  NE, 0,       262144 },
        /*nn  */ {16, 46, 27, 32, 64, hx_n, nuc_v,  NN, 1048576, 1114112},
        /*en  */ {20, 38, 25, 30, 60, hx_e, elec_v, NN, 1048576, 1114112},
    };

    for (int i = 0; i < 5; ++i) {
        const TD& t = tds[i];
        const float* dist = (const float*)d_in[t.in + 0];
        const float* dirs = (const float*)d_in[t.in + 1];
        const int*   send = (const int*)  d_in[t.in + 2];
        const int*   recv = (const int*)  d_in[t.in + 3];
        const int E = in_sizes[t.in] / 32;

        const float* Ww  = (const float*)d_in[t.w + 0];   // [32,384]
        const float* bw  = (const float*)d_in[t.w + 1];
        const float* Wg1 = (const float*)d_in[t.g + 0];   // [256,314]
        const float* bg1 = (const float*)d_in[t.g + 1];
        const float* Wg2 = (const float*)d_in[t.g + 2];   // [314,384]
        const float* bg2 = (const float*)d_in[t.g + 3];   // [384]
        const float* Vm  = (const float*)d_in[t.V];       // [128,128]
        const float* Um  = (const float*)d_in[t.U];       // [128,128]

        // repack this type's weights (cheap, stream-ordered)
        pack(Ww, 32, 384, Wwp, 32, 384);
        pack(Vm, 128, 128, Vp, 128, 128);
        pack(Um, 128, 128, Up, 128, 128);
        pack(Wg1, 256, 314, Wg1p, 256, 320);
        pad_vec<<<1, 320, 0, stream>>>(bg1, 314, bg1p, 320);
        pack(Wg2, 314, 384, Wg2p, 320, 384);

        // z_s and z_v are contiguous in ws -> single memset
        hipMemsetAsync(zs, 0, (size_t)(262144 + 786432) * 4, stream);

        const int groups = E / 16;
        edge_kernel<<<(groups + 7) / 8, 256, 0, stream>>>(
            dist, dirs, send, recv, Wwp, bw, t.hx, t.sv, zs, zv, E);

        // Vv/Uv = einsum('nci,cd->ndi'): 3 strided GEMMs each via gridDim.z
        gemmV(zv, Vp, Vv, t.Nr);
        gemmV(zv, Up, Uv, t.Nr);

        norm_concat<<<(t.Nr * 128 + 255) / 256, 256, 0, stream>>>(zs, Vv, gcat, t.Nr);

        gemmD(gcat, 256, Wg1p, 320, bg1p, t2, 320, t.Nr, 256, 1);
        gemmD(t2, 320, Wg2p, 384, bg2, gout, 384, t.Nr, 320, 0);

        finalize_add<<<(t.Nr * 128 + 255) / 256, 256, 0, stream>>>(
            gout, Uv, Vv, out + t.outS, out + t.outV, t.Nr);
    }
}